// FullLatentDistribution_8177617731697
// MI455X (gfx1250) — compile-verified
//
#include <hip/hip_runtime.h>
#include <hip/hip_bf16.h>

typedef __attribute__((ext_vector_type(2))) float v2f;
typedef __attribute__((ext_vector_type(8))) float v8f;

#define FLD_DIM   128
#define FLD_NP    8384        // 128 + 128*129/2
#define FLD_LSTR  132         // padded LDS row stride (floats): conflict-free b64 loads

__global__ __launch_bounds__(256) void fld_kernel(
    const float* __restrict__ params,   // [N, 8384]
    const float* __restrict__ eps,      // [N, 128]
    float* __restrict__ sample,         // [N, 128]
    float* __restrict__ kl,             // [N]
    float* __restrict__ Sigma)          // [N, 128, 128]
{
    const int n = blockIdx.x;
    const int t = threadIdx.x;

    const float* p = params + (size_t)n * FLD_NP;
    const float* x = p + FLD_DIM;       // packed tri data, 8256 elems

    __shared__ float Ls[FLD_DIM * FLD_LSTR];   // 67,584 B
    __shared__ float eps_s[FLD_DIM];
    __shared__ float mean_s[FLD_DIM];
    __shared__ float logd_s[FLD_DIM];
    __shared__ float klp_s[FLD_DIM];

    // ---- Build L in LDS -------------------------------------------------
    // xc[m] = (m < 8128) ? x[128+m] : x[16383-m], L = tril(xc.reshape(128,128))
    // off-diag rows scaled by 1/sqrt(i+1); diagonal replaced by exp(logdiag).
    for (int f = t; f < FLD_DIM * FLD_DIM; f += 256) {
        const int i = f >> 7;
        const int j = f & 127;
        float v = 0.0f;
        if (j <= i) {
            const float xv = (f < 8128) ? x[128 + f] : x[16383 - f];
            if (j == i) {
                logd_s[i] = xv;
                v = __expf(xv);
            } else {
                v = xv / sqrtf((float)(i + 1));
            }
        }
        Ls[i * FLD_LSTR + j] = v;
    }
    if (t < FLD_DIM) {
        mean_s[t] = p[t];
        eps_s[t]  = eps[(size_t)n * FLD_DIM + t];
    }
    __syncthreads();

    // ---- Sigma = L * L^T via V_WMMA_F32_16X16X4_F32 ---------------------
    // 32-bit A (16x4) lane map: lane l -> row = l%16, k-pair = 2*(l>>4).
    // For Sigma = L*L^T the B fragment of tile column tn equals the A
    // fragment of tile row tn, so both come from the same LDS addressing.
    const int wave = t >> 5;            // 0..7  -> tile row tm
    const int lane = t & 31;
    const int l16  = lane & 15;
    const int hh   = lane >> 4;         // 0/1: which k-pair / which M-half

    const float* arow = &Ls[(wave * 16 + l16) * FLD_LSTR + 2 * hh];
    v2f afrag[32];
    #pragma unroll
    for (int k4 = 0; k4 < 32; ++k4)
        afrag[k4] = *(const v2f*)(arow + 4 * k4);   // ds_load_b64, bank-conflict-free

    float* Sn = Sigma + (size_t)n * FLD_DIM * FLD_DIM;
    for (int tn = 0; tn < 8; ++tn) {
        const float* brow = &Ls[(tn * 16 + l16) * FLD_LSTR + 2 * hh];
        v8f acc = {};
        #pragma unroll
        for (int k4 = 0; k4 < 32; ++k4) {
            const v2f b = *(const v2f*)(brow + 4 * k4);
            acc = __builtin_amdgcn_wmma_f32_16x16x4_f32(
                /*neg_a=*/false, afrag[k4],
                /*neg_b=*/false, b,
                /*c_mod=*/(short)0, acc,
                /*reuse_a=*/false, /*reuse_b=*/false);
        }
        // C/D layout: lane l, vgpr v -> Sigma[tm*16 + v + 8*(l>>4)][tn*16 + l%16]
        #pragma unroll
        for (int v = 0; v < 8; ++v) {
            const int row = wave * 16 + v + 8 * hh;
            Sn[row * FLD_DIM + tn * 16 + l16] = acc[v];
        }
    }

    // ---- sample = mean + L*eps ; kl ------------------------------------
    if (t < FLD_DIM) {
        float acc = 0.0f, sq = 0.0f;
        const float* row = &Ls[t * FLD_LSTR];
        for (int j = 0; j <= t; ++j) {
            const float v = row[j];
            acc = fmaf(v, eps_s[j], acc);
            sq  = fmaf(v, v, sq);
        }
        sample[(size_t)n * FLD_DIM + t] = mean_s[t] + acc;
        const float m = mean_s[t];
        klp_s[t] = sq - 1.0f + m * m - 2.0f * logd_s[t];
    }
    __syncthreads();
    if (t == 0) {
        float s = 0.0f;
        for (int j = 0; j < FLD_DIM; ++j) s += klp_s[j];
        kl[n] = 0.5f * s;
    }
}

extern "C" void kernel_launch(void* const* d_in, const int* in_sizes, int n_in,
                              void* d_out, int out_size, void* d_ws, size_t ws_size,
                              hipStream_t stream) {
    const float* params = (const float*)d_in[0];   // [N, 8384] fp32
    const float* eps    = (const float*)d_in[1];   // [N, 128, 1] fp32
    const int N = in_sizes[0] / FLD_NP;            // 8192

    float* out    = (float*)d_out;
    float* sample = out;                                   // N*128
    float* kl     = out + (size_t)N * FLD_DIM;             // N
    float* Sigma  = kl + N;                                // N*128*128

    fld_kernel<<<N, 256, 0, stream>>>(params, eps, sample, kl, Sigma);
}